// Att_4_layer1_additive_save_memory_11879879544280
// MI455X (gfx1250) — compile-verified
//
#include <hip/hip_runtime.h>
#include <hip/hip_bf16.h>

typedef __attribute__((ext_vector_type(2))) float v2f;
typedef __attribute__((ext_vector_type(8))) float v8f;

__device__ __forceinline__ float fast_tanh(float x) {
#if __has_builtin(__builtin_amdgcn_tanhf)
    return __builtin_amdgcn_tanhf(x);
#else
    return tanhf(x);
#endif
}

// out[M,H] = A[M,D] * W[H,D]^T + bias[H]
// One wave32 per 16x16 output tile, K-step of 4 using V_WMMA_F32_16X16X4_F32.
// A-operand layout (ISA 7.12.2, 32-bit A 16x4): lane = half*16 + r holds
//   VGPR0 = A[r][k0+2*half], VGPR1 = A[r][k0+2*half+1].
// B-operand (4x16) assumed mirrored: lane holds W[h0+r][k0+2*half +{0,1}]
//   (contiguous float2 from the row-major weight -> coalesced-ish b64 loads).
// C/D layout: VGPR j -> M = j + 8*half, N = lane&15.
__global__ __launch_bounds__(128) void proj_gemm_wmma(
    const float* __restrict__ A,    // [M, D]
    const float* __restrict__ W,    // [H, D]
    const float* __restrict__ bias, // [H]
    float* __restrict__ out,        // [M, H]
    int M, int D, int H)
{
    const int wavesPerBlock = 4; // blockDim.x == 128
    int waveId = blockIdx.x * wavesPerBlock + (threadIdx.x >> 5);
    int tilesH = H >> 4;
    int tiles  = (M >> 4) * tilesH;
    if (waveId >= tiles) return;

    int m0 = (waveId / tilesH) << 4;
    int h0 = (waveId % tilesH) << 4;

    int lane = threadIdx.x & 31;
    int half = lane >> 4;   // 0 or 1
    int r    = lane & 15;

    const float* arow = A + (size_t)(m0 + r) * D + 2 * half;
    const float* wrow = W + (size_t)(h0 + r) * D + 2 * half;

    v8f acc = {};
#pragma unroll 4
    for (int k0 = 0; k0 < D; k0 += 4) {
        v2f a = *(const v2f*)(arow + k0);
        v2f b = *(const v2f*)(wrow + k0);
        acc = __builtin_amdgcn_wmma_f32_16x16x4_f32(
            /*neg_a=*/false, a, /*neg_b=*/false, b,
            /*c_mod=*/(short)0, acc, /*reuse_a=*/false, /*reuse_b=*/false);
    }

    float bcol = bias[h0 + r];
#pragma unroll
    for (int j = 0; j < 8; ++j) {
        int m = m0 + j + 8 * half;
        out[(size_t)m * H + h0 + r] = acc[j] + bcol;
    }
}

// One block (8 wave32) per (b,n,s) row.
//   logits[k] = sum_h Wl[h] * tanh(vp[b,k,h] + qp[row,h]) + bl
// followed by masked softmax over K boxes. Never materializes joint_repr.
__global__ __launch_bounds__(256) void joint_score_softmax(
    const float* __restrict__ vp,       // [B*K, H]
    const float* __restrict__ qp,       // [B*N*S, H]
    const float* __restrict__ Wl,       // [H]
    const float* __restrict__ bl,       // [1]
    const int*   __restrict__ box_mask, // [B, K]
    float* __restrict__ out)            // [B*N*S, K]
{
    const int H = 512, K = 50, NS = 4 * 64;
    int row  = blockIdx.x;          // 0..4095
    int b    = row / NS;
    int lane = threadIdx.x & 31;
    int wave = threadIdx.x >> 5;    // 0..7

    __shared__ float lds_logits[64];

    // per-lane strided slices of q_proj row and Wl (coalesced loads)
    float qv[16], wl[16];
#pragma unroll
    for (int j = 0; j < 16; ++j) {
        qv[j] = qp[(size_t)row * H + lane + 32 * j];
        wl[j] = Wl[lane + 32 * j];
    }
    float blv = bl[0];

    for (int k = wave; k < K; k += 8) {
        const float* vrow = vp + (size_t)(b * K + k) * H;
        float partial = 0.f;
#pragma unroll
        for (int j = 0; j < 16; ++j) {
            float t = fast_tanh(vrow[lane + 32 * j] + qv[j]);
            partial = __builtin_fmaf(wl[j], t, partial);
        }
#pragma unroll
        for (int off = 16; off > 0; off >>= 1)
            partial += __shfl_xor(partial, off, 32);
        if (lane == 0) lds_logits[k] = partial + blv;
    }
    __syncthreads();

    if (wave == 0) {
        // masked softmax over K=50 (2 elements per lane)
        int k2 = lane + 32;
        float l0 = -1e9f, l1 = -1e9f;
        if (lane < K) l0 = (box_mask[b * K + lane] > 0) ? lds_logits[lane] : -1e9f;
        if (k2 < K)  l1 = (box_mask[b * K + k2]  > 0) ? lds_logits[k2]  : -1e9f;

        float m = fmaxf(l0, l1);
#pragma unroll
        for (int off = 16; off > 0; off >>= 1)
            m = fmaxf(m, __shfl_xor(m, off, 32));

        float e0 = (lane < K) ? __expf(l0 - m) : 0.f;
        float e1 = (k2 < K)  ? __expf(l1 - m) : 0.f;
        float s = e0 + e1;
#pragma unroll
        for (int off = 16; off > 0; off >>= 1)
            s += __shfl_xor(s, off, 32);

        float inv = 1.f / s;
        if (lane < K) out[(size_t)row * K + lane] = e0 * inv;
        if (k2 < K)  out[(size_t)row * K + k2]  = e1 * inv;
    }
}

extern "C" void kernel_launch(void* const* d_in, const int* in_sizes, int n_in,
                              void* d_out, int out_size, void* d_ws, size_t ws_size,
                              hipStream_t stream) {
    // setup_inputs() order: v, q, box_mask, tags_attention(unused), Wv, bv, Wq, bq, Wl, bl
    const float* v        = (const float*)d_in[0]; // [16,1,50,1024] -> [800,1024]
    const float* q        = (const float*)d_in[1]; // [16,4,64,1,768] -> [4096,768]
    const int*   box_mask = (const int*)  d_in[2]; // [16,50]
    const float* Wv       = (const float*)d_in[4]; // [512,1024]
    const float* bv       = (const float*)d_in[5]; // [512]
    const float* Wq       = (const float*)d_in[6]; // [512,768]
    const float* bq       = (const float*)d_in[7]; // [512]
    const float* Wl       = (const float*)d_in[8]; // [512]
    const float* bl       = (const float*)d_in[9]; // [1]

    const int Mv = 800, Dv = 1024;   // B*K rows, VD
    const int Mq = 4096, Dq = 768;   // B*N*S rows, QD
    const int H = 512;

    float* vp = (float*)d_ws;              // [800, 512]
    float* qp = vp + (size_t)Mv * H;       // [4096, 512]

    {
        int tiles = (Mv / 16) * (H / 16);  // 1600
        proj_gemm_wmma<<<dim3((tiles + 3) / 4), 128, 0, stream>>>(v, Wv, bv, vp, Mv, Dv, H);
    }
    {
        int tiles = (Mq / 16) * (H / 16);  // 8192
        proj_gemm_wmma<<<dim3((tiles + 3) / 4), 128, 0, stream>>>(q, Wq, bq, qp, Mq, Dq, H);
    }
    joint_score_softmax<<<4096, 256, 0, stream>>>(vp, qp, Wl, bl, box_mask, (float*)d_out);
}